// GATREG_11261404250323
// MI455X (gfx1250) — compile-verified
//
#include <hip/hip_runtime.h>
#include <math.h>

#define NN 50000
#define EE 800000
#define DDIM 64
#define HHEADS 2
#define GGRAPHS 512

typedef float v2f __attribute__((ext_vector_type(2)));
typedef float v8f __attribute__((ext_vector_type(8)));

// ---------------------------------------------------------------- fill
__global__ __launch_bounds__(256) void fill_kernel(float* __restrict__ p, float v, int n) {
    int i = blockIdx.x * 256 + threadIdx.x;
    if (i < n) p[i] = v;
}

// ---------------------------------------------------------------- GEMM via V_WMMA_F32_16X16X4_F32
// Out[N x WIDTH] = A[N x 64] @ W[64 x WIDTH] + bias[WIDTH]
// One wave computes a 16-row x 64-col strip: A fragments loaded once into
// registers, 4 column tiles x 16 WMMA steps (independent chains interleave).
// WIDTH is a compile-time constant so all B/Out addressing folds to
// base + immediate offsets (no 64-bit mul in the loop).
template <int WIDTH>
__global__ __launch_bounds__(32) void gemm_strip_wmma(const float* __restrict__ A,
                                                      const float* __restrict__ W,
                                                      const float* __restrict__ bias,
                                                      float* __restrict__ Out) {
    const int tm   = blockIdx.x;               // 16-row strip index
    const int lane = threadIdx.x;              // 0..31 (wave32)
    const int r    = lane & 15;
    const int hi   = lane >> 4;                // 0: K pair {0,1}, 1: K pair {2,3}
    const int row  = tm * 16 + r;

    // Load all 16 A fragments for this strip once (32 VGPRs).
    v2f a[16];
    const float* __restrict__ arow = A + (size_t)row * DDIM + hi * 2;
    #pragma unroll
    for (int kk = 0; kk < 16; ++kk) {
        a[kk].x = arow[kk * 4];
        a[kk].y = arow[kk * 4 + 1];
    }

    const int tn0 = blockIdx.y * 4;            // 4 column tiles per wave
    #pragma unroll
    for (int t = 0; t < 4; ++t) {
        const int col = (tn0 + t) * 16 + r;
        const float* __restrict__ wb = W + (size_t)(hi * 2) * WIDTH + col;
        v8f acc = {0.f, 0.f, 0.f, 0.f, 0.f, 0.f, 0.f, 0.f};
        #pragma unroll
        for (int kk = 0; kk < 16; ++kk) {
            v2f b;
            b.x = wb[kk * 4 * WIDTH];          // K = 4*kk + 2*hi
            b.y = wb[kk * 4 * WIDTH + WIDTH];  // K = 4*kk + 2*hi + 1
            acc = __builtin_amdgcn_wmma_f32_16x16x4_f32(
                      false, a[kk], false, b, (short)0, acc, false, false);
        }
        const float bv = bias[col];
        float* __restrict__ op = Out + (size_t)(tm * 16 + hi * 8) * WIDTH + col;
        #pragma unroll
        for (int j = 0; j < 8; ++j)            // C/D: VGPR j -> M=j / M=j+8
            op[(size_t)j * WIDTH] = acc[j] + bv;
    }
}

// ---------------------------------------------------------------- per-edge scores + segment max
__device__ __forceinline__ void atomicMaxF(float* addr, float val) {
    if (val >= 0.f) atomicMax((int*)addr, __float_as_int(val));
    else            atomicMin((unsigned int*)addr, __float_as_uint(val));
}

__global__ __launch_bounds__(256) void edge_scores(const float* __restrict__ q,
                                                   const float* __restrict__ k,
                                                   const int* __restrict__ ei,
                                                   float* __restrict__ sc,
                                                   float* __restrict__ m) {
    int idx = blockIdx.x * 256 + threadIdx.x;
    if (idx >= EE * HHEADS) return;
    int e = idx >> 1, h = idx & 1;
    int s = ei[e], d = ei[EE + e];
    const float4* qp = (const float4*)(q + (size_t)d * 128 + h * 64);
    const float4* kp = (const float4*)(k + (size_t)s * 128 + h * 64);
    float sum = 0.f;
    #pragma unroll
    for (int c = 0; c < 16; ++c) {
        float4 a = qp[c], b = kp[c];
        sum += a.x * b.x + a.y * b.y + a.z * b.z + a.w * b.w;
    }
    sum *= 0.125f;                                   // 1/sqrt(64)
    sc[idx] = sum;
    atomicMaxF(&m[(size_t)d * HHEADS + h], sum);
}

// ---------------------------------------------------------------- exp + segment sum
__global__ __launch_bounds__(256) void edge_exp(const int* __restrict__ ei,
                                                float* __restrict__ sc,
                                                const float* __restrict__ m,
                                                float* __restrict__ denom) {
    int idx = blockIdx.x * 256 + threadIdx.x;
    if (idx >= EE * HHEADS) return;
    int e = idx >> 1, h = idx & 1;
    int d = ei[EE + e];
    float ev = __expf(sc[idx] - m[(size_t)d * HHEADS + h]);
    sc[idx] = ev;
    atomicAdd(&denom[(size_t)d * HHEADS + h], ev);
}

// ---------------------------------------------------------------- alpha * v scatter-add
__global__ __launch_bounds__(256) void edge_agg(const int* __restrict__ ei,
                                                const float* __restrict__ sc,
                                                const float* __restrict__ denom,
                                                const float* __restrict__ v,
                                                float* __restrict__ agg) {
    int idx = blockIdx.x * 256 + threadIdx.x;        // E * 32 threads, float4 per thread
    if (idx >= EE * 32) return;
    int e  = idx >> 5;
    int c4 = (idx & 31) * 4;                         // channel base 0..124
    int h  = c4 >> 6;
    int s = ei[e], d = ei[EE + e];
    float alpha = sc[(size_t)e * HHEADS + h] / (denom[(size_t)d * HHEADS + h] + 1e-16f);
    float4 vv = *(const float4*)(v + (size_t)s * 128 + c4);
    float* ap = agg + (size_t)d * 128 + c4;
    atomicAdd(ap + 0, alpha * vv.x);
    atomicAdd(ap + 1, alpha * vv.y);
    atomicAdd(ap + 2, alpha * vv.z);
    atomicAdd(ap + 3, alpha * vv.w);
}

// ---------------------------------------------------------------- head-mean + skip + relu (+ layernorm)
__global__ __launch_bounds__(256) void combine_ln(const float* __restrict__ agg,
                                                  const float* __restrict__ skip,
                                                  const float* __restrict__ lg,
                                                  const float* __restrict__ lb,
                                                  float* __restrict__ out,
                                                  int do_ln) {
    int lane = threadIdx.x & 31;
    int n = blockIdx.x * 8 + (threadIdx.x >> 5);     // 8 waves per block, 1 node per wave
    if (n >= NN) return;
    const float* ap = agg + (size_t)n * 128;
    const float* sp = skip + (size_t)n * 64;
    int c0 = lane, c1 = lane + 32;
    float v0 = 0.5f * (ap[c0] + ap[64 + c0]) + sp[c0];
    float v1 = 0.5f * (ap[c1] + ap[64 + c1]) + sp[c1];
    v0 = fmaxf(v0, 0.f);
    v1 = fmaxf(v1, 0.f);
    if (do_ln) {
        float s = v0 + v1;
        for (int off = 16; off > 0; off >>= 1) s += __shfl_xor(s, off, 32);
        float mu = s * (1.f / 64.f);
        float d0 = v0 - mu, d1 = v1 - mu;
        float qq = d0 * d0 + d1 * d1;
        for (int off = 16; off > 0; off >>= 1) qq += __shfl_xor(qq, off, 32);
        float inv = rsqrtf(qq * (1.f / 64.f) + 1e-5f);
        v0 = d0 * inv * lg[c0] + lb[c0];
        v1 = d1 * inv * lg[c1] + lb[c1];
    }
    float* op = out + (size_t)n * 64;
    op[c0] = v0;
    op[c1] = v1;
}

// ---------------------------------------------------------------- graph mean-pool accumulate
__global__ __launch_bounds__(256) void pool_kernel(const float* __restrict__ h,
                                                   const int* __restrict__ batch,
                                                   float* __restrict__ psum,
                                                   float* __restrict__ pcnt) {
    int idx = blockIdx.x * 256 + threadIdx.x;
    if (idx >= NN * 64) return;
    int n = idx >> 6, c = idx & 63;
    int g = batch[n];
    atomicAdd(&psum[(size_t)g * 64 + c], h[idx]);
    if (c == 0) atomicAdd(&pcnt[g], 1.f);
}

// ---------------------------------------------------------------- final projection
__global__ __launch_bounds__(32) void final_kernel(const float* __restrict__ psum,
                                                   const float* __restrict__ pcnt,
                                                   const float* __restrict__ Wout,
                                                   const float* __restrict__ bout,
                                                   float* __restrict__ out) {
    int g = blockIdx.x;
    int lane = threadIdx.x;
    float inv = 1.f / fmaxf(pcnt[g], 1.f);
    float s = psum[(size_t)g * 64 + lane] * inv * Wout[lane]
            + psum[(size_t)g * 64 + lane + 32] * inv * Wout[lane + 32];
    for (int off = 16; off > 0; off >>= 1) s += __shfl_xor(s, off, 32);
    if (lane == 0) out[g] = s + bout[0];
}

// ================================================================ launch
extern "C" void kernel_launch(void* const* d_in, const int* in_sizes, int n_in,
                              void* d_out, int out_size, void* d_ws, size_t ws_size,
                              hipStream_t stream) {
    const float* x    = (const float*)d_in[0];
    const int*   ei   = (const int*)d_in[1];     // [2, E]: src row then dst row
    const int*   bat  = (const int*)d_in[2];
    const float* Wq   = (const float*)d_in[4];
    const float* bq   = (const float*)d_in[5];
    const float* Wk   = (const float*)d_in[6];
    const float* bk   = (const float*)d_in[7];
    const float* Wv   = (const float*)d_in[8];
    const float* bv   = (const float*)d_in[9];
    const float* Ws   = (const float*)d_in[10];
    const float* bs   = (const float*)d_in[11];
    const float* lng  = (const float*)d_in[12];
    const float* lnb  = (const float*)d_in[13];
    const float* Wout = (const float*)d_in[14];
    const float* bout = (const float*)d_in[15];
    float* out = (float*)d_out;

    float* ws    = (float*)d_ws;
    float* q     = ws;
    float* kbuf  = q     + (size_t)NN * 128;
    float* vbuf  = kbuf  + (size_t)NN * 128;
    float* skip  = vbuf  + (size_t)NN * 128;
    float* h0    = skip  + (size_t)NN * 64;
    float* h1    = h0    + (size_t)NN * 64;
    float* sc    = h1    + (size_t)NN * 64;
    float* m     = sc    + (size_t)EE * HHEADS;
    float* denom = m     + (size_t)NN * HHEADS;
    float* agg   = denom + (size_t)NN * HHEADS;
    float* psum  = agg   + (size_t)NN * 128;
    float* pcnt  = psum  + (size_t)GGRAPHS * 64;

    hipMemcpyAsync(h0, x, sizeof(float) * (size_t)NN * 64, hipMemcpyDeviceToDevice, stream);

    float* hin = h0;
    float* hout = h1;
    const dim3 gQKV(NN / 16, 128 / 64);   // 3125 x 2 waves (4 col tiles each)
    const dim3 gS(NN / 16, 64 / 64);      // 3125 x 1
    const int nEH   = EE * HHEADS;        // 1.6M
    const int bEH   = (nEH + 255) / 256;
    const int nAGG  = EE * 32;            // 25.6M
    const int bAGG  = (nAGG + 255) / 256;

    for (int l = 0; l < 3; ++l) {
        gemm_strip_wmma<128><<<gQKV, 32, 0, stream>>>(hin, Wq + (size_t)l * 64 * 128, bq + l * 128, q);
        gemm_strip_wmma<128><<<gQKV, 32, 0, stream>>>(hin, Wk + (size_t)l * 64 * 128, bk + l * 128, kbuf);
        gemm_strip_wmma<128><<<gQKV, 32, 0, stream>>>(hin, Wv + (size_t)l * 64 * 128, bv + l * 128, vbuf);
        gemm_strip_wmma<64><<<gS,   32, 0, stream>>>(hin, Ws + (size_t)l * 64 * 64,  bs + l * 64,  skip);

        fill_kernel<<<(NN * HHEADS + 255) / 256, 256, 0, stream>>>(m, -INFINITY, NN * HHEADS);
        fill_kernel<<<(NN * HHEADS + 255) / 256, 256, 0, stream>>>(denom, 0.f, NN * HHEADS);
        fill_kernel<<<(NN * 128 + 255) / 256, 256, 0, stream>>>(agg, 0.f, NN * 128);

        edge_scores<<<bEH, 256, 0, stream>>>(q, kbuf, ei, sc, m);
        edge_exp<<<bEH, 256, 0, stream>>>(ei, sc, m, denom);
        edge_agg<<<bAGG, 256, 0, stream>>>(ei, sc, denom, vbuf, agg);

        int do_ln = (l < 2) ? 1 : 0;
        int li = (l < 2) ? l : 1;   // keep pointer in-bounds even when LN disabled
        combine_ln<<<(NN + 7) / 8, 256, 0, stream>>>(agg, skip, lng + li * 64, lnb + li * 64, hout, do_ln);

        float* t = hin; hin = hout; hout = t;
    }

    fill_kernel<<<(GGRAPHS * 64 + GGRAPHS + 255) / 256, 256, 0, stream>>>(psum, 0.f, GGRAPHS * 64 + GGRAPHS);
    pool_kernel<<<(NN * 64 + 255) / 256, 256, 0, stream>>>(hin, bat, psum, pcnt);
    final_kernel<<<GGRAPHS, 32, 0, stream>>>(psum, pcnt, Wout, bout, out);
}